// LU_45853070852237
// MI455X (gfx1250) — compile-verified
//
#include <hip/hip_runtime.h>
#include <hip/hip_bf16.h>
#include <cstdint>

// Batched 9x(5x5) block-sparse LU, B = 524288 independent batch elements.
// Memory-bound: ~944 MB total traffic -> ~40us floor @ 23.3 TB/s.
// Strategy: each 64-thread workgroup owns 64 batch elements (57600 B).
//   1) coalesced async DMA global->LDS (global_load_async_to_lds_b128),
//      one base VGPR + instruction `offset:` immediates (added to both the
//      global and LDS addresses per the ISA) -> no per-op address VALU.
//   2) one thread = one batch element, LU + fill-in computed from LDS
//      (batch stride 225 dwords, odd mod 64 -> bank-conflict free)
//   3) coalesced async DMA LDS->global (global_store_async_from_lds_b128)

#define WG    64            // threads per block == batches per block
#define ELEMS 225           // floats per batch element (9*5*5)
#define BPB   (WG * 900)    // bytes per workgroup tile = 57600
#define LDSF  (WG * ELEMS)  // 14400 floats = 57600 B of LDS

// Unpivoted 5x5 LU, in-place on 25 contiguous floats (row-major), staged
// through registers. Matches reference: col = A[k+1:,k] / A[k,k]; Schur update.
__device__ __forceinline__ void lu5_inplace(float* a) {
    float m[25];
#pragma unroll
    for (int i = 0; i < 25; ++i) m[i] = a[i];
#pragma unroll
    for (int k = 0; k < 5; ++k) {
        const float p = m[k * 5 + k];
#pragma unroll
        for (int i = k + 1; i < 5; ++i) m[i * 5 + k] = m[i * 5 + k] / p;
#pragma unroll
        for (int i = k + 1; i < 5; ++i)
#pragma unroll
            for (int j = k + 1; j < 5; ++j)
                m[i * 5 + j] -= m[i * 5 + k] * m[k * 5 + j];
    }
#pragma unroll
    for (int i = 0; i < 25; ++i) a[i] = m[i];
}

__global__ __launch_bounds__(WG) void lu_blocksparse_kernel(
    const float* __restrict__ in, float* __restrict__ out, long long nBatch) {
    __shared__ float smem[LDSF];

    const int tid            = (int)threadIdx.x;
    const long long blk      = (long long)blockIdx.x;
    const long long baseByte = blk * (long long)BPB;
    const long long totalB   = nBatch * 900ll;
    const long long remB     = totalB - baseByte;
    const int nbytes         = (int)(remB < (long long)BPB ? remB : (long long)BPB);

    const char* gin  = (const char*)in + baseByte;        // uniform -> SGPR pair
    char*       gout = (char*)out + baseByte;             // uniform -> SGPR pair
    const unsigned lds0 = (unsigned)(uintptr_t)&smem[0];  // LDS byte offset

    // ---------------- async DMA: global -> LDS (coalesced) ----------------
    if (nbytes == BPB) {
        // Fast path (always taken for B % 64 == 0): single base address pair,
        // 56 x b128 rounds via the 24-bit instruction offset (added to BOTH
        // the global and the LDS address), plus one b32 round for the last
        // 256 B of the 57600 B tile (57600 = 56*64*16 + 64*4).
        const unsigned g16 = (unsigned)(tid * 16);
        const unsigned l16 = lds0 + g16;
#pragma unroll
        for (int r = 0; r < 56; ++r) {
            asm volatile("global_load_async_to_lds_b128 %0, %1, %2 offset:%3"
                         :: "v"(l16), "v"(g16), "s"(gin), "i"(r * 1024)
                         : "memory");
        }
        const unsigned g4 = (unsigned)(56 * WG * 16 + tid * 4);
        asm volatile("global_load_async_to_lds_b32 %0, %1, %2 offset:0"
                     :: "v"(lds0 + g4), "v"(g4), "s"(gin) : "memory");
    } else {
        // Ragged last tile: per-op guards with dword fallback.
        for (int r = 0; r < 56; ++r) {
            const unsigned off = (unsigned)((r * WG + tid) * 16);
            if ((int)off + 16 <= nbytes) {
                asm volatile("global_load_async_to_lds_b128 %0, %1, %2 offset:0"
                             :: "v"(lds0 + off), "v"(off), "s"(gin) : "memory");
            } else {
#pragma unroll
                for (int q = 0; q < 4; ++q) {
                    const unsigned o2 = off + (unsigned)(q * 4);
                    if ((int)o2 + 4 <= nbytes)
                        asm volatile("global_load_async_to_lds_b32 %0, %1, %2 offset:0"
                                     :: "v"(lds0 + o2), "v"(o2), "s"(gin) : "memory");
                }
            }
        }
        const unsigned off = (unsigned)(56 * WG * 16 + tid * 4);
        if ((int)off + 4 <= nbytes)
            asm volatile("global_load_async_to_lds_b32 %0, %1, %2 offset:0"
                         :: "v"(lds0 + off), "v"(off), "s"(gin) : "memory");
    }
    asm volatile("s_wait_asynccnt 0" ::: "memory");
    __syncthreads();

    // ---------------- per-thread LU + fill-in, all in LDS ----------------
    if (blk * WG + tid < nBatch) {
        float* A = &smem[tid * ELEMS];
        // layer 0: LU blocks 0,1,2,5,6
        lu5_inplace(A + 0 * 25);
        lu5_inplace(A + 1 * 25);
        lu5_inplace(A + 2 * 25);
        // fill-in layer0 -> block 3 (dst -= src); (b,r,c) flat = b*25+r*5+c
        A[3 * 25 + 0] -= A[1 * 25 + 6] + A[2 * 25 + 12]; // (1,1,1)+(2,2,2) -> (3,0,0)
        A[3 * 25 + 1] -= A[2 * 25 + 13];                 // (2,2,3) -> (3,0,1)
        A[3 * 25 + 5] -= A[2 * 25 + 17];                 // (2,3,2) -> (3,1,0)
        A[3 * 25 + 6] -= A[2 * 25 + 18];                 // (2,3,3) -> (3,1,1)
        lu5_inplace(A + 3 * 25);
        lu5_inplace(A + 5 * 25);
        lu5_inplace(A + 6 * 25);
        // fill-in layer0 -> block 7
        A[7 * 25 + 0] -= A[5 * 25 + 6] + A[6 * 25 + 18]; // (5,1,1)+(6,3,3) -> (7,0,0)
        A[7 * 25 + 1] -= A[6 * 25 + 19];                 // (6,3,4) -> (7,0,1)
        A[7 * 25 + 5] -= A[6 * 25 + 23];                 // (6,4,3) -> (7,1,0)
        A[7 * 25 + 6] -= A[6 * 25 + 24];                 // (6,4,4) -> (7,1,1)
        lu5_inplace(A + 7 * 25);
        // fill-in layer1 -> block 8
        A[8 * 25 + 0] -= A[0 * 25 + 6] + A[3 * 25 + 6] + A[7 * 25 + 6];
        lu5_inplace(A + 8 * 25);
        // block 4 is untouched by the reference: passes through LDS verbatim.
    }
    __syncthreads();

    // ---------------- async DMA: LDS -> global (coalesced) ----------------
    if (nbytes == BPB) {
        const unsigned g16 = (unsigned)(tid * 16);
        const unsigned l16 = lds0 + g16;
#pragma unroll
        for (int r = 0; r < 56; ++r) {
            asm volatile("global_store_async_from_lds_b128 %0, %1, %2 offset:%3"
                         :: "v"(g16), "v"(l16), "s"(gout), "i"(r * 1024)
                         : "memory");
        }
        const unsigned g4 = (unsigned)(56 * WG * 16 + tid * 4);
        asm volatile("global_store_async_from_lds_b32 %0, %1, %2 offset:0"
                     :: "v"(g4), "v"(lds0 + g4), "s"(gout) : "memory");
    } else {
        for (int r = 0; r < 56; ++r) {
            const unsigned off = (unsigned)((r * WG + tid) * 16);
            if ((int)off + 16 <= nbytes) {
                asm volatile("global_store_async_from_lds_b128 %0, %1, %2 offset:0"
                             :: "v"(off), "v"(lds0 + off), "s"(gout) : "memory");
            } else {
#pragma unroll
                for (int q = 0; q < 4; ++q) {
                    const unsigned o2 = off + (unsigned)(q * 4);
                    if ((int)o2 + 4 <= nbytes)
                        asm volatile("global_store_async_from_lds_b32 %0, %1, %2 offset:0"
                                     :: "v"(o2), "v"(lds0 + o2), "s"(gout) : "memory");
                }
            }
        }
        const unsigned off = (unsigned)(56 * WG * 16 + tid * 4);
        if ((int)off + 4 <= nbytes)
            asm volatile("global_store_async_from_lds_b32 %0, %1, %2 offset:0"
                         :: "v"(off), "v"(lds0 + off), "s"(gout) : "memory");
    }
    asm volatile("s_wait_asynccnt 0" ::: "memory");
    // S_ENDPGM performs an implicit wait-idle; stores are committed at kernel end.
}

extern "C" void kernel_launch(void* const* d_in, const int* in_sizes, int n_in,
                              void* d_out, int out_size, void* d_ws, size_t ws_size,
                              hipStream_t stream) {
    const float* in  = (const float*)d_in[0];
    float*       out = (float*)d_out;
    const long long totalElems = (long long)in_sizes[0];  // B * 9 * 5 * 5
    const long long nBatch     = totalElems / ELEMS;      // 524288
    const int nBlocks          = (int)((nBatch + WG - 1) / WG); // 8192
    lu_blocksparse_kernel<<<nBlocks, WG, 0, stream>>>(in, out, nBatch);
}